// RNN_38448547233991
// MI455X (gfx1250) — compile-verified
//
#include <hip/hip_runtime.h>
#include <cmath>

typedef __attribute__((ext_vector_type(2))) float v2f;
typedef __attribute__((ext_vector_type(8))) float v8f;

#define EMBED 300
#define HID 20
#define NCLS 15
#define B_SZ 128
#define S_LEN 512
#define KSTEPS_E (EMBED / 4) /* 75 */
#define KSTEPS_H (HID / 4)   /* 5  */

#if __has_builtin(__builtin_amdgcn_tanhf)
#define TANHF(x) __builtin_amdgcn_tanhf(x)
#else
#define TANHF(x) tanhf(x)
#endif

// ---------------------------------------------------------------------------
// Kernel 1: fused embedding gather + input projection (WMMA f32 GEMM)
//   xw[s][b][n] = sum_e emb[X[b*S+s]][e] * W_ih[n][e] + b_ih[n]
//   GEMM view: M = B*S = 65536 token rows, K = 300, N = 20 (padded to 32).
//   One wave computes a 16(M) x 32(N) tile. A rows stream from the gathered
//   embedding rows (single-use -> prefetch + non-temporal); B operands are
//   staged once per block in LDS in exact WMMA B layout.
// ---------------------------------------------------------------------------
__global__ __launch_bounds__(256) void rnn_proj_kernel(
    const int* __restrict__ X,       // [B*S] token ids
    const float* __restrict__ emb,   // [VOCAB][300]
    const float* __restrict__ W_ih,  // [20][300]
    const float* __restrict__ b_ih,  // [20]
    float* __restrict__ xw)          // [S][B][20]
{
  __shared__ v2f Wlds[KSTEPS_E * 2 * 32]; // 75 k-steps x 2 N-tiles x 32 lanes = 38.4 KB

  const int tid = threadIdx.x;

  // Stage W_ih^T into LDS in B-operand layout:
  //   lane l holds (B[k][col], B[k+1][col]), col = tile*16 + (l&15),
  //   k = 4*kk + 2*(l>>4);  B[k][col] = W_ih[col][k]; cols >= 20 zero-padded.
  for (int e = tid; e < KSTEPS_E * 2 * 32; e += 256) {
    const int lane = e & 31;
    const int tl   = (e >> 5) & 1;
    const int kk   = e >> 6;
    const int col  = tl * 16 + (lane & 15);
    const int k    = kk * 4 + 2 * (lane >> 4);
    v2f w = {};
    if (col < HID) {
      w.x = W_ih[col * EMBED + k];
      w.y = W_ih[col * EMBED + k + 1];
    }
    Wlds[e] = w;
  }
  __syncthreads();

  const int lane = tid & 31;
  const int wv   = tid >> 5;
  const int r    = lane & 15;
  const int hi   = lane >> 4;

  const int mtile = blockIdx.x * 8 + wv;   // 16-row M tile owned by this wave
  const int m_row = mtile * 16 + r;        // this lane's A row
  const long long tok = X[m_row];
  const float* rowbase = emb + (size_t)tok * EMBED;

  // Kick the whole 1200-byte row toward the caches before the k-loop.
#pragma unroll
  for (int p = 0; p < 5; ++p)
    __builtin_prefetch(rowbase + p * 64, 0, 1);

  const float* arow = rowbase + 2 * hi;

  v8f c0 = {}; // N cols 0..15
  v8f c1 = {}; // N cols 16..31 (16..19 valid)
  for (int kk = 0; kk < KSTEPS_E; ++kk) {
    const v2f a  = __builtin_nontemporal_load((const v2f*)(arow + kk * 4));
    const v2f b0 = Wlds[(kk * 2 + 0) * 32 + lane];
    const v2f b1 = Wlds[(kk * 2 + 1) * 32 + lane];
    c0 = __builtin_amdgcn_wmma_f32_16x16x4_f32(false, a, false, b0, (short)0, c0, false, false);
    c1 = __builtin_amdgcn_wmma_f32_16x16x4_f32(false, a, false, b1, (short)0, c1, false, false);
  }

  // D layout: VGPR j -> (M = j + 8*hi, N = r [+16 for tile1]).
  // A 16-row tile never crosses a batch boundary (16 | 512), so one base
  // pointer + immediate offsets (j * 2560 floats = 10240 B < 2^23).
  const int m0 = mtile * 16;
  const int b  = m0 >> 9;             // S_LEN == 512
  const int s0 = m0 & (S_LEN - 1);
  float* obase = xw + ((size_t)s0 + 8 * hi) * (B_SZ * HID) + b * HID;

  const float bias0 = b_ih[r];
  float* p0 = obase + r;
#pragma unroll
  for (int j = 0; j < 8; ++j)
    p0[(size_t)j * (B_SZ * HID)] = c0[j] + bias0;
  if (r < 4) {
    const float bias1 = b_ih[16 + r];
    float* p1 = obase + 16 + r;
#pragma unroll
    for (int j = 0; j < 8; ++j)
      p1[(size_t)j * (B_SZ * HID)] = c1[j] + bias1;
  }
}

// ---------------------------------------------------------------------------
// Kernel 2: sequential recurrence + fused classifier head.
//   h_t = tanh(xw_t + h_{t-1} @ W_hh^T + b_hh);  out = h_S @ W_out^T + b_out
//   Batch rows are independent: 8 blocks x 1 wave, each wave owns 16 batch
//   rows on its own WGP. h tile lives in LDS (LDS bounce = D->A transpose),
//   W_hh^T B-operands live in registers, 10 f32 WMMAs per step, no barriers
//   (same-wave DS ops are in-order).
// ---------------------------------------------------------------------------
__global__ __launch_bounds__(32) void rnn_scan_kernel(
    const float* __restrict__ h0,    // [128][20]
    const float* __restrict__ W_hh,  // [20][20]
    const float* __restrict__ b_hh,  // [20]
    const float* __restrict__ xw,    // [512][128][20]
    const float* __restrict__ W_out, // [15][20]
    const float* __restrict__ b_out, // [15]
    float* __restrict__ out)         // [128][15]
{
  __shared__ float hw[16 * HID];     // this wave's 16x20 h tile (1.28 KB)

  const int lane = threadIdx.x & 31;
  const int wv   = blockIdx.x;       // batch tile: rows [wv*16, wv*16+16)
  const int r    = lane & 15;
  const int hi   = lane >> 4;

  for (int e = lane; e < 16 * HID; e += 32)
    hw[e] = h0[wv * 16 * HID + e];

  // Preload B = W_hh^T operands (time-invariant): B[k][col] = W_hh[col][k].
  v2f bop[KSTEPS_H][2];
#pragma unroll
  for (int kk = 0; kk < KSTEPS_H; ++kk) {
#pragma unroll
    for (int tl = 0; tl < 2; ++tl) {
      const int col = tl * 16 + r;
      const int k   = kk * 4 + 2 * hi;
      v2f b = {};
      if (col < HID) { b.x = W_hh[col * HID + k]; b.y = W_hh[col * HID + k + 1]; }
      bop[kk][tl] = b;
    }
  }
  const float bias0 = b_hh[r];
  const float bias1 = (r < 4) ? b_hh[16 + r] : 0.f;

  for (int t = 0; t < S_LEN; ++t) {
    // A operands from LDS: lane holds h[row=r][4*kk + 2*hi .. +1].
    v2f a[KSTEPS_H];
#pragma unroll
    for (int kk = 0; kk < KSTEPS_H; ++kk)
      a[kk] = *(const v2f*)(hw + r * HID + kk * 4 + 2 * hi);

    v8f c0 = {}; v8f c1 = {};
#pragma unroll
    for (int kk = 0; kk < KSTEPS_H; ++kk) {
      c0 = __builtin_amdgcn_wmma_f32_16x16x4_f32(false, a[kk], false, bop[kk][0], (short)0, c0, false, false);
      c1 = __builtin_amdgcn_wmma_f32_16x16x4_f32(false, a[kk], false, bop[kk][1], (short)0, c1, false, false);
    }

    const float* xrow = xw + (size_t)t * (B_SZ * HID) + (wv * 16 + 8 * hi) * HID;
#pragma unroll
    for (int j = 0; j < 8; ++j) {
      const int m = j + 8 * hi;       // local row 0..15
      hw[m * HID + r] = TANHF(c0[j] + bias0 + xrow[j * HID + r]);
      if (r < 4)
        hw[m * HID + 16 + r] = TANHF(c1[j] + bias1 + xrow[j * HID + 16 + r]);
    }
    // Same-wave LDS ops are in-order: next iteration's loads see these stores.
  }

  // Fused head on the final hidden state (still resident in LDS).
  for (int e = lane; e < 16 * NCLS; e += 32) {
    const int lb = e / NCLS, c = e % NCLS;
    float acc = b_out[c];
#pragma unroll
    for (int k = 0; k < HID; ++k)
      acc += hw[lb * HID + k] * W_out[c * HID + k];
    out[(wv * 16 + lb) * NCLS + c] = acc;
  }
}

extern "C" void kernel_launch(void* const* d_in, const int* in_sizes, int n_in,
                              void* d_out, int out_size, void* d_ws, size_t ws_size,
                              hipStream_t stream) {
  const int*   X     = (const int*)  d_in[0];
  const float* h0    = (const float*)d_in[1];
  const float* emb   = (const float*)d_in[2];
  const float* W_ih  = (const float*)d_in[3];
  const float* b_ih  = (const float*)d_in[4];
  const float* W_hh  = (const float*)d_in[5];
  const float* b_hh  = (const float*)d_in[6];
  const float* W_out = (const float*)d_in[7];
  const float* b_out = (const float*)d_in[8];
  float* out = (float*)d_out;

  float* xw = (float*)d_ws; // [512][128][20] f32 = 5.24 MB (L2-resident)

  // 65536 token rows / (16 rows/wave * 8 waves/block) = 512 blocks
  rnn_proj_kernel<<<dim3((B_SZ * S_LEN) / (16 * 8)), 256, 0, stream>>>(X, emb, W_ih, b_ih, xw);
  // 8 batch tiles, one wave each, spread across WGPs
  rnn_scan_kernel<<<dim3(8), 32, 0, stream>>>(h0, W_hh, b_hh, xw, W_out, b_out, out);
}